// DGL_24653112279736
// MI455X (gfx1250) — compile-verified
//
#include <hip/hip_runtime.h>
#include <hip/hip_bf16.h>

typedef __attribute__((ext_vector_type(2))) float v2f;
typedef __attribute__((ext_vector_type(8))) float v8f;

#define B_SZ 4
#define F_SZ 64
#define N_SZ 4096
#define T_SZ 12
#define DIM  768            // T*F
#define DK   32
#define K_SELECT 1677721    // int(4096*4096*0.1)
#define SA   770            // padded LDS row stride (floats): 770 % 64 == 2 -> conflict-free b64 lanes
#define SCORE_SCALE 0.17677669529663687f  // 1/sqrt(32)
#define DROP_THRESH 0.1f
#define DROP_SCALE  (1.0f / 0.9f)

__device__ __forceinline__ v8f wmma4(v2f a, v2f b, v8f c) {
    // D = A(16x4,f32) * B(4x16,f32) + C(16x16,f32)
    return __builtin_amdgcn_wmma_f32_16x16x4_f32(
        /*neg_a=*/false, a, /*neg_b=*/false, b,
        /*c_mod=*/(short)0, c, /*reuse_a=*/false, /*reuse_b=*/false);
}

// CDNA5 async global->LDS copy (no VGPR round trip, tracked by ASYNCcnt).
// Generic pointers into LDS carry the LDS byte offset in their low 32 bits.
__device__ __forceinline__ void async_g2l_b32(void* ldsPtr, const void* gPtr) {
    unsigned ldsOff = (unsigned)(unsigned long long)ldsPtr;
    unsigned long long ga = (unsigned long long)gPtr;
    asm volatile("global_load_async_to_lds_b32 %0, %1, off"
                 :: "v"(ldsOff), "v"(ga) : "memory");
}
__device__ __forceinline__ void wait_async0() {
    asm volatile("s_wait_asynccnt 0x0" ::: "memory");
}

// ---------------------------------------------------------------------------
// Kernel 1: fused Q/K projection.  Grid = B * (N/16) blocks, 128 thr (4 waves).
// LDS: A-tile 16x768 (gathered transpose of x) + both W matrices transposed,
// staged with GLOBAL_LOAD_ASYNC_TO_LDS_B32.
// Wave w computes cols [16w,16w+16) of the concatenated [Q|K] 64-col output.
// ---------------------------------------------------------------------------
__global__ void proj_kernel(const float* __restrict__ x,
                            const float* __restrict__ WQ,
                            const float* __restrict__ WK,
                            float* __restrict__ Qout,
                            float* __restrict__ Kout) {
    extern __shared__ float smem[];
    float* ldsA = smem;             // 16 rows x SA
    float* ldsW = smem + 16 * SA;   // 64 cols x SA (cols 0-31 = WQ, 32-63 = WK)

    const int blk = blockIdx.x;           // 0..1023
    const int b   = blk >> 8;             // 256 row-tiles per batch
    const int n0  = (blk & 255) << 4;
    const int tid = threadIdx.x;          // 128 threads

    const float* xb = x + (size_t)b * (F_SZ * N_SZ * T_SZ);

    // Stage W transposed: ldsW[col][c] = W[c][col]   (async, ASYNCcnt)
    for (int idx = tid; idx < 64 * DIM; idx += 128) {
        int c  = idx >> 6;
        int cc = idx & 63;
        const float* src = (cc < DK) ? (WQ + c * DK + cc) : (WK + c * DK + (cc - DK));
        async_g2l_b32(&ldsW[cc * SA + c], src);
    }
    // Stage A: xr[n0+row][t*64+f] = x[b][f][n0+row][t]   (async, ASYNCcnt)
    for (int idx = tid; idx < 16 * DIM; idx += 128) {
        int f   = idx / 192;
        int w   = idx - f * 192;
        int row = w / T_SZ;
        int t   = w - row * T_SZ;
        async_g2l_b32(&ldsA[row * SA + t * F_SZ + f],
                      xb + f * (N_SZ * T_SZ) + (n0 + row) * T_SZ + t);
    }
    wait_async0();        // s_wait_asynccnt 0
    __syncthreads();

    const int lane = tid & 31;
    const int wave = tid >> 5;
    const int l16  = lane & 15;
    const int half = lane >> 4;
    const int colBase = wave << 4;

    const float* aRow = ldsA + l16 * SA + half * 2;
    const float* bRow = ldsW + (colBase + l16) * SA + half * 2;

    v8f acc = {};
#pragma unroll 4
    for (int k0 = 0; k0 < DIM; k0 += 4) {
        v2f a  = *(const v2f*)(aRow + k0);   // ds_load_b64
        v2f bb = *(const v2f*)(bRow + k0);   // ds_load_b64
        acc = wmma4(a, bb, acc);
    }

    const int col = colBase + l16;           // uniform-per-wave branch
    float* outp = (col < DK)
        ? (Qout + (size_t)b * (N_SZ * DK) + col)
        : (Kout + (size_t)b * (N_SZ * DK) + (col - DK));
#pragma unroll
    for (int v = 0; v < 8; ++v) {
        int row = v + half * 8;
        outp[(size_t)(n0 + row) * DK] = acc[v];
    }
}

// ---------------------------------------------------------------------------
// Kernel 2: S[n,m] = (1/sqrt(32)) * sum_b Q[b,n,:].K[b,m,:]
// One 16x16 tile per wave; 8 waves/block; 8192 blocks. Batch sum folded into
// the WMMA accumulator (32 wmmas per tile).
// ---------------------------------------------------------------------------
__global__ void score_kernel(const float* __restrict__ Q,
                             const float* __restrict__ K,
                             float* __restrict__ S) {
    const int tid  = threadIdx.x;
    const int lane = tid & 31;
    const int wave = tid >> 5;
    const int tile = blockIdx.x * 8 + wave;   // 0..65535
    const int n0   = (tile >> 8) << 4;
    const int m0   = (tile & 255) << 4;
    const int l16  = lane & 15;
    const int half = lane >> 4;

    v8f acc = {};
#pragma unroll
    for (int b = 0; b < B_SZ; ++b) {
        const float* aP = Q + (size_t)b * (N_SZ * DK) + (n0 + l16) * DK + half * 2;
        const float* bP = K + (size_t)b * (N_SZ * DK) + (m0 + l16) * DK + half * 2;
        __builtin_prefetch(bP, 0, 0);   // global_prefetch_b8 on the K stream
#pragma unroll
        for (int k0 = 0; k0 < DK; k0 += 4) {
            v2f a  = *(const v2f*)(aP + k0);   // global_load_b64
            v2f bb = *(const v2f*)(bP + k0);   // global_load_b64
            acc = wmma4(a, bb, acc);
        }
    }
#pragma unroll
    for (int v = 0; v < 8; ++v) {
        int row = n0 + v + half * 8;
        S[(size_t)row * N_SZ + m0 + l16] = acc[v] * SCORE_SCALE;
    }
}

// ---------------------------------------------------------------------------
// Kernel 3: row softmax in place, fused with radix-select pass-0 histogram
// (saves one full 64 MB read of S).  One block (256 thr) per row.
// ---------------------------------------------------------------------------
__global__ void softmax_hist_kernel(float* __restrict__ S,
                                    int* __restrict__ hist1) {
    __shared__ float red[256];
    __shared__ int h[4096];
    const int row = blockIdx.x;
    const int tid = threadIdx.x;
    float* p = S + (size_t)row * N_SZ;

    for (int i = tid; i < 4096; i += 256) h[i] = 0;

    float r[16];
    float mx = -3.402823466e38f;
#pragma unroll
    for (int i = 0; i < 16; ++i) {
        r[i] = p[tid + (i << 8)];
        mx = fmaxf(mx, r[i]);
    }
    red[tid] = mx;
    __syncthreads();
    for (int s = 128; s > 0; s >>= 1) {
        if (tid < s) red[tid] = fmaxf(red[tid], red[tid + s]);
        __syncthreads();
    }
    mx = red[0];
    __syncthreads();

    float sum = 0.f;
#pragma unroll
    for (int i = 0; i < 16; ++i) {
        r[i] = __expf(r[i] - mx);    // v_exp_f32
        sum += r[i];
    }
    red[tid] = sum;
    __syncthreads();
    for (int s = 128; s > 0; s >>= 1) {
        if (tid < s) red[tid] += red[tid + s];
        __syncthreads();
    }
    const float inv = 1.0f / red[0];
#pragma unroll
    for (int i = 0; i < 16; ++i) {
        float v = r[i] * inv;
        p[tid + (i << 8)] = v;
        atomicAdd(&h[__float_as_uint(v) >> 20], 1);   // pass-0 radix histogram
    }
    __syncthreads();
    for (int i = tid; i < 4096; i += 256)
        if (h[i]) atomicAdd(&hist1[i], h[i]);
}

// ---------------------------------------------------------------------------
// Global top-k (exact) via 3-pass radix select on float bit patterns.
// Softmax outputs are >= 0 so uint bit order == float order.
// ---------------------------------------------------------------------------
__global__ void zero_kernel(int* __restrict__ p, int n) {
    int i = blockIdx.x * blockDim.x + threadIdx.x;
    if (i < n) p[i] = 0;
}

__global__ void hist_kernel(const float* __restrict__ S,
                            int* __restrict__ hist,
                            const int* __restrict__ state,
                            int pass) {
    __shared__ int h[4096];
    const int nb = (pass == 2) ? 256 : 4096;
    for (int i = threadIdx.x; i < nb; i += blockDim.x) h[i] = 0;
    __syncthreads();

    int filt = 0;
    if (pass == 1)      filt = state[0];
    else if (pass == 2) filt = (state[0] << 12) | state[2];

    const int total = N_SZ * N_SZ;
    for (int i = blockIdx.x * blockDim.x + threadIdx.x; i < total;
         i += gridDim.x * blockDim.x) {
        unsigned u = __float_as_uint(S[i]);
        if (pass == 1) {
            if ((int)(u >> 20) == filt) atomicAdd(&h[(u >> 8) & 0xFFF], 1);
        } else {
            if ((int)(u >> 8) == filt) atomicAdd(&h[u & 0xFF], 1);
        }
    }
    __syncthreads();
    for (int i = threadIdx.x; i < nb; i += blockDim.x)
        if (h[i]) atomicAdd(&hist[i], h[i]);
}

__global__ void select_kernel(const int* __restrict__ hist,
                              int* __restrict__ state, int pass) {
    if (threadIdx.x != 0 || blockIdx.x != 0) return;
    const int nb  = (pass == 2) ? 256 : 4096;
    const int kin = (pass == 0) ? K_SELECT : state[pass * 2 - 1];
    int acc = 0, bsel = 0, krem = kin;
    for (int b = nb - 1; b >= 0; --b) {
        int c = hist[b];
        acc += c;
        if (acc >= kin) { bsel = b; krem = kin - (acc - c); break; }
    }
    state[pass * 2]     = bsel;
    state[pass * 2 + 1] = krem;
    if (pass == 2) {
        state[6] = (int)(((unsigned)state[0] << 20) |
                         ((unsigned)state[2] << 8) | (unsigned)bsel);
        state[7] = krem;   // ties needed at exact threshold value
        state[8] = 0;      // tie counter
    }
}

// ---------------------------------------------------------------------------
// Final: apply top-k mask + dropout on the remainder, in place.
// ---------------------------------------------------------------------------
__global__ void drop_kernel(float* __restrict__ S,
                            const float* __restrict__ du,
                            int* __restrict__ state) {
    const unsigned thr = (unsigned)state[6];
    const int needed   = state[7];
    int* tie = state + 8;
    const int total = N_SZ * N_SZ;
    for (int i = blockIdx.x * blockDim.x + threadIdx.x; i < total;
         i += gridDim.x * blockDim.x) {
        float v    = S[i];
        unsigned u = __float_as_uint(v);
        bool kept;
        if (u > thr)       kept = true;
        else if (u == thr) kept = (atomicAdd(tie, 1) < needed);
        else               kept = false;
        if (!kept) S[i] = (du[i] >= DROP_THRESH) ? v * DROP_SCALE : 0.0f;
    }
}

extern "C" void kernel_launch(void* const* d_in, const int* in_sizes, int n_in,
                              void* d_out, int out_size, void* d_ws, size_t ws_size,
                              hipStream_t stream) {
    const float* x  = (const float*)d_in[0];   // [4,64,4096,12]
    const float* WQ = (const float*)d_in[1];   // [768,32]
    const float* WK = (const float*)d_in[2];   // [768,32]
    const float* du = (const float*)d_in[3];   // [4096,4096]
    float* S = (float*)d_out;                  // [4096,4096] — also scratch

    float* Q = (float*)d_ws;                   // [4,4096,32]
    float* K = Q + B_SZ * N_SZ * DK;           // [4,4096,32]
    int* ints  = (int*)(K + B_SZ * N_SZ * DK);
    int* hist1 = ints;                         // 4096
    int* hist2 = hist1 + 4096;                 // 4096
    int* hist3 = hist2 + 4096;                 // 256
    int* state = hist3 + 256;                  // 16

    zero_kernel<<<(8464 + 255) / 256, 256, 0, stream>>>(ints, 8464);

    const size_t lds1 = (size_t)(16 + 64) * SA * sizeof(float);  // ~246 KB (<320 KB/WGP)
    proj_kernel<<<B_SZ * (N_SZ / 16), 128, lds1, stream>>>(x, WQ, WK, Q, K);

    score_kernel<<<(N_SZ / 16) * (N_SZ / 16) / 8, 256, 0, stream>>>(Q, K, S);

    softmax_hist_kernel<<<N_SZ, 256, 0, stream>>>(S, hist1);
    select_kernel<<<1, 32, 0, stream>>>(hist1, state, 0);
    hist_kernel<<<2048, 256, 0, stream>>>(S, hist2, state, 1);
    select_kernel<<<1, 32, 0, stream>>>(hist2, state, 1);
    hist_kernel<<<2048, 256, 0, stream>>>(S, hist3, state, 2);
    select_kernel<<<1, 32, 0, stream>>>(hist3, state, 2);

    drop_kernel<<<2048, 256, 0, stream>>>(S, du, state);
}